// SGDAT_40638980555010
// MI455X (gfx1250) — compile-verified
//
#include <hip/hip_runtime.h>
#include <math.h>

typedef __attribute__((ext_vector_type(2))) float v2f;
typedef __attribute__((ext_vector_type(8))) float v8f;

#define BATCH 8
#define NPTS 100000
#define NP1 512
#define NP2 128

__device__ __forceinline__ v8f wmma4(v2f a, v2f b, v8f c) {
  // D = A(16x4) * B(4x16) + C, full fp32 (exact vs reference)
  return __builtin_amdgcn_wmma_f32_16x16x4_f32(false, a, false, b, (short)0, c, false, false);
}

// ---------------------------------------------------------------- utilities

__global__ void zero_kernel(float* __restrict__ p, int n) {
  int i = blockIdx.x * blockDim.x + threadIdx.x;
  if (i < n) p[i] = 0.f;
}

// Per-batch xyz mean / (1/std, ddof=1, clipped at 1e-6). stats[b*6+c]=mean, [b*6+3+c]=rstd
__global__ __launch_bounds__(256) void stats_kernel(const float* __restrict__ pts,
                                                    float* __restrict__ stats, int N) {
  int b = blockIdx.x, tid = threadIdx.x;
  const float* P = pts + (size_t)b * N * 9;
  double s0 = 0, s1 = 0, s2 = 0, q0 = 0, q1 = 0, q2 = 0;
  for (int i = tid; i < N; i += 256) {
    double x = P[(size_t)i * 9 + 0], y = P[(size_t)i * 9 + 1], z = P[(size_t)i * 9 + 2];
    s0 += x; s1 += y; s2 += z; q0 += x * x; q1 += y * y; q2 += z * z;
  }
  __shared__ double sh[256][6];
  sh[tid][0] = s0; sh[tid][1] = s1; sh[tid][2] = s2;
  sh[tid][3] = q0; sh[tid][4] = q1; sh[tid][5] = q2;
  __syncthreads();
  for (int st = 128; st > 0; st >>= 1) {
    if (tid < st)
      for (int j = 0; j < 6; ++j) sh[tid][j] += sh[tid + st][j];
    __syncthreads();
  }
  if (tid < 3) {
    double mean = sh[0][tid] / N;
    double var = (sh[0][3 + tid] - (double)N * mean * mean) / (double)(N - 1);
    double sd = sqrt(var > 0.0 ? var : 0.0);
    if (sd < 1e-6) sd = 1e-6;
    stats[b * 6 + tid] = (float)mean;
    stats[b * 6 + 3 + tid] = (float)(1.0 / sd);
  }
}

// Farthest point sampling: one 1024-thread workgroup per batch; dist kept in L2.
// Records idx + (normalized) xyz of each selected point. Matches reference: first idx = 0.
__global__ __launch_bounds__(1024) void fps_kernel(const float* __restrict__ X, int stride,
                                                   const float* __restrict__ stats,
                                                   float* __restrict__ dist,
                                                   int* __restrict__ idx_out,
                                                   float* __restrict__ xyz_out, int N, int np) {
  int b = blockIdx.x, tid = threadIdx.x;
  const float* P = X + (size_t)b * N * stride;
  const float* st = stats ? stats + b * 6 : nullptr;
  float* D = dist + (size_t)b * N;
  __shared__ float rv[1024];
  __shared__ int ri[1024];
  for (int i = tid; i < N; i += 1024) D[i] = 1e10f;
  __syncthreads();
  int far = 0;
  for (int it = 0; it < np; ++it) {
    float c0 = P[(size_t)far * stride + 0];
    float c1 = P[(size_t)far * stride + 1];
    float c2 = P[(size_t)far * stride + 2];
    if (st) {
      c0 = (c0 - st[0]) * st[3]; c1 = (c1 - st[1]) * st[4]; c2 = (c2 - st[2]) * st[5];
    }
    if (tid == 0) {
      idx_out[b * np + it] = far;
      xyz_out[((size_t)b * np + it) * 3 + 0] = c0;
      xyz_out[((size_t)b * np + it) * 3 + 1] = c1;
      xyz_out[((size_t)b * np + it) * 3 + 2] = c2;
    }
    float bv = -1.f;
    int bi = 0;
    for (int i = tid; i < N; i += 1024) {
      float x = P[(size_t)i * stride + 0];
      float y = P[(size_t)i * stride + 1];
      float z = P[(size_t)i * stride + 2];
      if (st) {
        x = (x - st[0]) * st[3]; y = (y - st[1]) * st[4]; z = (z - st[2]) * st[5];
      }
      float dx = x - c0, dy = y - c1, dz = z - c2;
      float d = dx * dx + dy * dy + dz * dz;
      float dm = fminf(D[i], d);
      D[i] = dm;
      if (dm > bv) { bv = dm; bi = i; }
    }
    rv[tid] = bv; ri[tid] = bi;
    __syncthreads();
    for (int s = 512; s > 0; s >>= 1) {
      if (tid < s) {
        float ov = rv[tid + s]; int oi = ri[tid + s];
        if (ov > rv[tid] || (ov == rv[tid] && oi < ri[tid])) { rv[tid] = ov; ri[tid] = oi; }
      }
      __syncthreads();
    }
    far = ri[0];
    __syncthreads();
  }
}

// SE gate from channel sums: gate = sigmoid(relu(mean@w1+b1)@w2+b2).  blockDim == C.
__global__ void se_gate_kernel(const float* __restrict__ hsumv, const float* __restrict__ sw1,
                               const float* __restrict__ sb1, const float* __restrict__ sw2,
                               const float* __restrict__ sb2, float* __restrict__ gate, int C,
                               int H, float invM) {
  int b = blockIdx.x, c = threadIdx.x;
  __shared__ float mn[128];
  __shared__ float s1[16];
  mn[c] = hsumv[b * C + c] * invM;
  __syncthreads();
  if (c < H) {
    float a = sb1[c];
    for (int k = 0; k < C; ++k) a += mn[k] * sw1[k * H + c];
    s1[c] = fmaxf(a, 0.f);
  }
  __syncthreads();
  float a = sb2[c];
  for (int h = 0; h < H; ++h) a += s1[h] * sw2[h * C + c];
  gate[b * C + c] = 1.f / (1.f + expf(-a));
}

// ----------------------------------------------- enc0 (9 -> 64), N = 100000
// Pass 1: WMMA h2 = bn2(relu(bn1(x@w1))@w2); only accumulate per-channel sums (for SE mean).

__global__ __launch_bounds__(256) void enc0_sum_kernel(
    const float* __restrict__ pts, const float* __restrict__ stats, const float* __restrict__ w1,
    const float* __restrict__ g1, const float* __restrict__ b1, const float* __restrict__ w2,
    const float* __restrict__ g2, const float* __restrict__ b2, float* __restrict__ hsum0, int N,
    int tilesPerB) {
  __shared__ float xs[128][12];
  __shared__ float hs1[128][66];  // padded stride -> conflict-free ds_load_b64 A-frags
  __shared__ float csum[64];
  int b = blockIdx.x / tilesPerB;
  int base = (blockIdx.x % tilesPerB) * 128;
  int tid = threadIdx.x;
  if (tid < 64) csum[tid] = 0.f;
  __builtin_prefetch(w2, 0, 3);
  const float* P = pts + (size_t)b * N * 9;
  const float* st = stats + b * 6;
  for (int i = tid; i < 128 * 9; i += 256) {
    int m = i / 9, f = i % 9, p = base + m;
    float v = 0.f;
    if (p < N) {
      v = P[(size_t)p * 9 + f];
      if (f < 3) v = (v - st[f]) * st[3 + f];
    }
    xs[m][f] = v;
  }
  __syncthreads();
  const float rb = rsqrtf(1.001f);  // 1/sqrt(1+BN_EPS)
  for (int j = 0; j < 32; ++j) {
    int o = j * 256 + tid, m = o >> 6, c = o & 63;
    float a = 0.f;
#pragma unroll
    for (int k = 0; k < 9; ++k) a += xs[m][k] * w1[k * 64 + c];
    a = a * (g1[c] * rb) + b1[c];
    hs1[m][c] = fmaxf(a, 0.f);
  }
  __syncthreads();
  int wave = tid >> 5, lane = tid & 31, half = lane >> 4, ln = lane & 15;
  int mt = wave * 16;  // 8 waves x 16 rows = 128 points
  v8f acc[4] = {};
  for (int ks = 0; ks < 16; ++ks) {
    int k = 4 * ks + 2 * half;
    v2f a = *(const v2f*)&hs1[mt + ln][k];
#pragma unroll
    for (int nt = 0; nt < 4; ++nt) {
      int n = nt * 16 + ln;
      v2f bb;
      bb.x = w2[k * 64 + n];
      bb.y = w2[(k + 1) * 64 + n];
      acc[nt] = wmma4(a, bb, acc[nt]);
    }
  }
  for (int nt = 0; nt < 4; ++nt) {
    int n = nt * 16 + ln;
    float sc = g2[n] * rb, bo = b2[n];
    float part = 0.f;
#pragma unroll
    for (int r = 0; r < 8; ++r) {
      int m = mt + r + 8 * half;
      if (base + m < N) part += acc[nt][r] * sc + bo;  // mask tail points
    }
    atomicAdd(&csum[n], part);
  }
  __syncthreads();
  if (tid < 64) atomicAdd(&hsum0[b * 64 + tid], csum[tid]);
}

// Pass 2: recompute enc0 only at the 512 FPS-gathered points, apply SE gate + residual,
// and emit f512_in = [xyz(3) | f0(64)] with row stride 72.
__global__ __launch_bounds__(256) void enc0_out_kernel(
    const float* __restrict__ pts, const float* __restrict__ stats, const int* __restrict__ idx512,
    const float* __restrict__ w1, const float* __restrict__ g1, const float* __restrict__ b1,
    const float* __restrict__ w2, const float* __restrict__ g2, const float* __restrict__ b2,
    const float* __restrict__ gate, const float* __restrict__ wproj, float* __restrict__ fin_out,
    int N) {
  __shared__ float xs[128][12];
  __shared__ float hs1[128][66];
  int b = blockIdx.x >> 2;
  int base = (blockIdx.x & 3) * 128;
  int tid = threadIdx.x;
  const float* P = pts + (size_t)b * N * 9;
  const float* st = stats + b * 6;
  for (int i = tid; i < 128 * 9; i += 256) {
    int m = i / 9, f = i % 9;
    int p = idx512[b * NP1 + base + m];
    float v = P[(size_t)p * 9 + f];
    if (f < 3) {
      v = (v - st[f]) * st[3 + f];
      fin_out[((size_t)b * NP1 + base + m) * 72 + f] = v;
    }
    xs[m][f] = v;
  }
  __syncthreads();
  const float rb = rsqrtf(1.001f);
  for (int j = 0; j < 32; ++j) {
    int o = j * 256 + tid, m = o >> 6, c = o & 63;
    float a = 0.f;
#pragma unroll
    for (int k = 0; k < 9; ++k) a += xs[m][k] * w1[k * 64 + c];
    a = a * (g1[c] * rb) + b1[c];
    hs1[m][c] = fmaxf(a, 0.f);
  }
  __syncthreads();
  int wave = tid >> 5, lane = tid & 31, half = lane >> 4, ln = lane & 15;
  int mt = wave * 16;
  v8f acc[4] = {};
  for (int ks = 0; ks < 16; ++ks) {
    int k = 4 * ks + 2 * half;
    v2f a = *(const v2f*)&hs1[mt + ln][k];
#pragma unroll
    for (int nt = 0; nt < 4; ++nt) {
      int n = nt * 16 + ln;
      v2f bb;
      bb.x = w2[k * 64 + n];
      bb.y = w2[(k + 1) * 64 + n];
      acc[nt] = wmma4(a, bb, acc[nt]);
    }
  }
  for (int nt = 0; nt < 4; ++nt) {
    int n = nt * 16 + ln;
    float sc = g2[n] * rb, bo = b2[n], gt = gate[b * 64 + n];
#pragma unroll
    for (int r = 0; r < 8; ++r) {
      int m = mt + r + 8 * half;
      float h = (acc[nt][r] * sc + bo) * gt;
      float res = 0.f;
#pragma unroll
      for (int k = 0; k < 9; ++k) res += xs[m][k] * wproj[k * 64 + n];
      fin_out[((size_t)b * NP1 + base + m) * 72 + 3 + n] = fmaxf(h + res, 0.f);
    }
  }
}

// ----------------------------- generic 67 -> COUT ResMLP "h" pass (enc512 / enc128)
// Computes h = bn2(relu(bn1(x@w1))@w2) for a 64-point tile, stores h and channel sums.
template <int COUT>
__global__ __launch_bounds__(256) void enc_h_kernel(
    const float* __restrict__ fin, int fstride, int Mtot, const float* __restrict__ w1,
    const float* __restrict__ g1, const float* __restrict__ b1, const float* __restrict__ w2,
    const float* __restrict__ g2, const float* __restrict__ b2, float* __restrict__ hbuf,
    float* __restrict__ hsumv) {
  constexpr int TILE = 64;
  __shared__ float xs[TILE][68];
  __shared__ float hs1[TILE][COUT + 2];
  __shared__ float csum[COUT];
  int tilesPerB = Mtot / TILE;
  int b = blockIdx.x / tilesPerB;
  int base = (blockIdx.x % tilesPerB) * TILE;
  int tid = threadIdx.x;
  if (tid < COUT) csum[tid] = 0.f;
  __builtin_prefetch(w2, 0, 3);
  const float* F = fin + ((size_t)b * Mtot + base) * fstride;
  for (int i = tid; i < TILE * 67; i += 256) {
    int m = i / 67, k = i % 67;
    xs[m][k] = F[(size_t)m * fstride + k];
  }
  __syncthreads();
  const float rb = rsqrtf(1.001f);
  constexpr int PER = TILE * COUT / 256;
  for (int j = 0; j < PER; ++j) {
    int o = j * 256 + tid, m = o / COUT, c = o % COUT;
    float a = 0.f;
    for (int k = 0; k < 67; ++k) a += xs[m][k] * w1[k * COUT + c];
    a = a * (g1[c] * rb) + b1[c];
    hs1[m][c] = fmaxf(a, 0.f);
  }
  __syncthreads();
  int wave = tid >> 5, lane = tid & 31, half = lane >> 4, ln = lane & 15;
  int mt = (wave & 3) * 16;            // 4 M-tiles cover the 64 points
  constexpr int NTW = COUT / 32;       // n-tiles per wave (2 for 64ch, 4 for 128ch)
  int ntb = (wave >> 2) * NTW;         // wave groups split the N dimension
  v8f acc[NTW] = {};
  for (int ks = 0; ks < COUT / 4; ++ks) {
    int k = 4 * ks + 2 * half;
    v2f a = *(const v2f*)&hs1[mt + ln][k];
#pragma unroll
    for (int j = 0; j < NTW; ++j) {
      int n = (ntb + j) * 16 + ln;
      v2f bb;
      bb.x = w2[k * COUT + n];
      bb.y = w2[(k + 1) * COUT + n];
      acc[j] = wmma4(a, bb, acc[j]);
    }
  }
  for (int j = 0; j < NTW; ++j) {
    int n = (ntb + j) * 16 + ln;
    float sc = g2[n] * rb, bo = b2[n];
    float part = 0.f;
#pragma unroll
    for (int r = 0; r < 8; ++r) {
      int m = mt + r + 8 * half;
      float h = acc[j][r] * sc + bo;
      hbuf[((size_t)b * Mtot + base + m) * COUT + n] = h;
      part += h;
    }
    atomicAdd(&csum[n], part);
  }
  __syncthreads();
  if (tid < COUT) atomicAdd(&hsumv[b * COUT + tid], csum[tid]);
}

// Epilogue: out = relu(h*gate + x@wproj).  One block per point, blockDim == COUT.
__global__ void enc_out_kernel(const float* __restrict__ fin, int fstride, int Mtot,
                               const float* __restrict__ hbuf, const float* __restrict__ gate,
                               const float* __restrict__ wproj, float* __restrict__ out, int COUT) {
  int p = blockIdx.x;  // b*Mtot + m
  int b = p / Mtot;
  int c = threadIdx.x;
  __shared__ float xf[67];
  for (int k = c; k < 67; k += blockDim.x) xf[k] = fin[(size_t)p * fstride + k];
  __syncthreads();
  float res = 0.f;
  for (int k = 0; k < 67; ++k) res += xf[k] * wproj[k * COUT + c];
  float h = hbuf[(size_t)p * COUT + c] * gate[b * COUT + c];
  out[(size_t)p * COUT + c] = fmaxf(h + res, 0.f);
}

// Build f128_in = [xyz128(3) | gather(f512)(64)] with row stride 72.
__global__ void gather_fin_kernel(const float* __restrict__ feat, int Min, int C,
                                  const float* __restrict__ xyz_s, const int* __restrict__ idx,
                                  int Mout, float* __restrict__ fin, int fstride) {
  int g = blockIdx.x;  // b*Mout + m
  int b = g / Mout, m = g % Mout;
  int src = idx[b * Mout + m];
  int tid = threadIdx.x;
  if (tid < 3) fin[(size_t)g * fstride + tid] = xyz_s[(size_t)g * 3 + tid];
  for (int c = tid; c < C; c += blockDim.x)
    fin[(size_t)g * fstride + 3 + c] = feat[((size_t)b * Min + src) * C + c];
}

// ---------------------------------------------------------------- launcher

extern "C" void kernel_launch(void* const* d_in, const int* in_sizes, int n_in, void* d_out,
                              int out_size, void* d_ws, size_t ws_size, hipStream_t stream) {
  (void)in_sizes; (void)n_in; (void)out_size; (void)ws_size;
  int i = 0;
  const float* pts = (const float*)d_in[i++];
  // p0 (9->64)
  const float* w1_0 = (const float*)d_in[i++]; const float* g1_0 = (const float*)d_in[i++];
  const float* b1_0 = (const float*)d_in[i++]; const float* w2_0 = (const float*)d_in[i++];
  const float* g2_0 = (const float*)d_in[i++]; const float* b2_0 = (const float*)d_in[i++];
  const float* sw1_0 = (const float*)d_in[i++]; const float* sb1_0 = (const float*)d_in[i++];
  const float* sw2_0 = (const float*)d_in[i++]; const float* sb2_0 = (const float*)d_in[i++];
  const float* wp_0 = (const float*)d_in[i++];
  // p512 (67->64)
  const float* w1_5 = (const float*)d_in[i++]; const float* g1_5 = (const float*)d_in[i++];
  const float* b1_5 = (const float*)d_in[i++]; const float* w2_5 = (const float*)d_in[i++];
  const float* g2_5 = (const float*)d_in[i++]; const float* b2_5 = (const float*)d_in[i++];
  const float* sw1_5 = (const float*)d_in[i++]; const float* sb1_5 = (const float*)d_in[i++];
  const float* sw2_5 = (const float*)d_in[i++]; const float* sb2_5 = (const float*)d_in[i++];
  const float* wp_5 = (const float*)d_in[i++];
  // p128 (67->128)
  const float* w1_1 = (const float*)d_in[i++]; const float* g1_1 = (const float*)d_in[i++];
  const float* b1_1 = (const float*)d_in[i++]; const float* w2_1 = (const float*)d_in[i++];
  const float* g2_1 = (const float*)d_in[i++]; const float* b2_1 = (const float*)d_in[i++];
  const float* sw1_1 = (const float*)d_in[i++]; const float* sb1_1 = (const float*)d_in[i++];
  const float* sw2_1 = (const float*)d_in[i++]; const float* sb2_1 = (const float*)d_in[i++];
  const float* wp_1 = (const float*)d_in[i++];

  // carve workspace (256B aligned slabs)
  char* base = (char*)d_ws;
  size_t off = 0;
  auto carve = [&](size_t bytes) -> void* {
    void* p = base + off;
    off = (off + bytes + 255) & ~(size_t)255;
    return p;
  };
  float* stats = (float*)carve(BATCH * 6 * 4);
  float* dist = (float*)carve((size_t)BATCH * NPTS * 4);
  int* idx512 = (int*)carve(BATCH * NP1 * 4);
  float* xyz512 = (float*)carve(BATCH * NP1 * 3 * 4);
  float* f512in = (float*)carve((size_t)BATCH * NP1 * 72 * 4);
  float* hsum0 = (float*)carve(BATCH * 64 * 4);
  float* gate0 = (float*)carve(BATCH * 64 * 4);
  float* hsum512 = (float*)carve(BATCH * 64 * 4);
  float* gate512 = (float*)carve(BATCH * 64 * 4);
  float* hbuf512 = (float*)carve((size_t)BATCH * NP1 * 64 * 4);
  float* f512 = (float*)carve((size_t)BATCH * NP1 * 64 * 4);
  int* idx128 = (int*)carve(BATCH * NP2 * 4);
  float* xyz128 = (float*)carve(BATCH * NP2 * 3 * 4);
  float* f128in = (float*)carve((size_t)BATCH * NP2 * 72 * 4);
  float* hsum128 = (float*)carve(BATCH * 128 * 4);
  float* gate128 = (float*)carve(BATCH * 128 * 4);
  float* hbuf128 = (float*)carve((size_t)BATCH * NP2 * 128 * 4);
  float* out = (float*)d_out;

  const int N = NPTS;
  const int tiles0 = (N + 127) / 128;

  stats_kernel<<<BATCH, 256, 0, stream>>>(pts, stats, N);
  zero_kernel<<<(BATCH * 64 + 255) / 256, 256, 0, stream>>>(hsum0, BATCH * 64);
  zero_kernel<<<(BATCH * 64 + 255) / 256, 256, 0, stream>>>(hsum512, BATCH * 64);
  zero_kernel<<<(BATCH * 128 + 255) / 256, 256, 0, stream>>>(hsum128, BATCH * 128);

  // FPS over normalized xyz (normalization applied inline from stats)
  fps_kernel<<<BATCH, 1024, 0, stream>>>(pts, 9, stats, dist, idx512, xyz512, N, NP1);

  // enc0: channel-sum pass (WMMA), SE gate, recompute at 512 gathered points
  enc0_sum_kernel<<<BATCH * tiles0, 256, 0, stream>>>(pts, stats, w1_0, g1_0, b1_0, w2_0, g2_0,
                                                      b2_0, hsum0, N, tiles0);
  se_gate_kernel<<<BATCH, 64, 0, stream>>>(hsum0, sw1_0, sb1_0, sw2_0, sb2_0, gate0, 64, 8,
                                           1.0f / (float)N);
  enc0_out_kernel<<<BATCH * 4, 256, 0, stream>>>(pts, stats, idx512, w1_0, g1_0, b1_0, w2_0, g2_0,
                                                 b2_0, gate0, wp_0, f512in, N);

  // second FPS level (512 -> 128), xyz512 already normalized
  fps_kernel<<<BATCH, 1024, 0, stream>>>(xyz512, 3, nullptr, dist, idx128, xyz128, NP1, NP2);

  // enc512: 67->64 ResMLP over 512 points
  enc_h_kernel<64><<<BATCH * (NP1 / 64), 256, 0, stream>>>(f512in, 72, NP1, w1_5, g1_5, b1_5, w2_5,
                                                           g2_5, b2_5, hbuf512, hsum512);
  se_gate_kernel<<<BATCH, 64, 0, stream>>>(hsum512, sw1_5, sb1_5, sw2_5, sb2_5, gate512, 64, 8,
                                           1.0f / (float)NP1);
  enc_out_kernel<<<BATCH * NP1, 64, 0, stream>>>(f512in, 72, NP1, hbuf512, gate512, wp_5, f512, 64);

  // gather f128_in = [xyz128 | f512[idx128]]
  gather_fin_kernel<<<BATCH * NP2, 64, 0, stream>>>(f512, NP1, 64, xyz128, idx128, NP2, f128in, 72);

  // enc128: 67->128 ResMLP over 128 points -> d_out
  enc_h_kernel<128><<<BATCH * (NP2 / 64), 256, 0, stream>>>(f128in, 72, NP2, w1_1, g1_1, b1_1, w2_1,
                                                            g2_1, b2_1, hbuf128, hsum128);
  se_gate_kernel<<<BATCH, 128, 0, stream>>>(hsum128, sw1_1, sb1_1, sw2_1, sb2_1, gate128, 128, 16,
                                            1.0f / (float)NP2);
  enc_out_kernel<<<BATCH * NP2, 128, 0, stream>>>(f128in, 72, NP2, hbuf128, gate128, wp_1, out, 128);
}